// to_boundary_1073741824326
// MI455X (gfx1250) — compile-verified
//
#include <hip/hip_runtime.h>

typedef __attribute__((ext_vector_type(2))) float v2f;
typedef __attribute__((ext_vector_type(4))) float v4f;
typedef __attribute__((ext_vector_type(8))) float v8f;
typedef __attribute__((__vector_size__(4 * sizeof(int)))) int v4i;
typedef __attribute__((address_space(3))) v4i as3_v4i;

#define KF  (-0.01f)            // F
#define KW  (-0.1f)             // W
#define C3  (2.0f*KW - KF)      // coefficient on box3 = 2W - F = -0.19
#define CD  (2.0f*(1.0f - KW))  // coefficient on delta = 2(H - W) = 2.2

#define TS     60   // output tile (stride-1 maxpool result)
#define ES     64   // edge region tile = TS + 4 (maxpool halo)
#define IS     68   // input tile = ES + 4 (conv halo)
#define LSTR   68   // LDS row stride (floats); 272 B -> 16-B aligned rows

// ---- CDNA5 async global->LDS path (ASYNCcnt), with safe fallback ----
#if defined(__has_builtin)
#if __has_builtin(__builtin_amdgcn_global_load_async_to_lds_b128)
#define HAS_ASYNC_G2L 1
#endif
#endif

#ifdef HAS_ASYNC_G2L
__device__ __forceinline__ void async_g2l_b128(const void* g, void* l) {
    __builtin_amdgcn_global_load_async_to_lds_b128((v4i*)g, (as3_v4i*)l, 0, 0);
}
#endif

__device__ __forceinline__ void wait_async0() {
#ifdef HAS_ASYNC_G2L
#if __has_builtin(__builtin_amdgcn_s_wait_asynccnt)
    __builtin_amdgcn_s_wait_asynccnt(0);
#else
    asm volatile("s_wait_asynccnt 0" ::: "memory");
#endif
#endif
}

__device__ __forceinline__ float max5(float a, float b, float c, float d, float e) {
    return fmaxf(fmaxf(fmaxf(a, b), fmaxf(c, d)), e);
}

__global__ __launch_bounds__(256) void edge_pool_kernel(const float* __restrict__ x,
                                                        float* __restrict__ out,
                                                        int nplanes)
{
    __shared__ float lds[3 * IS * LSTR];   // 55,488 bytes
    float* xs = lds;                // [68][68] input tile; later overlaid by edge[64][64]
    float* h5 = lds + IS * LSTR;    // [68][64] 5-wide row sums; later overlaid by rmax[64][60]
    float* h3 = lds + 2 * IS * LSTR;// [68][64] 3-wide row sums

    const int tid   = threadIdx.x;
    const int plane = blockIdx.z;
    const int row0  = blockIdx.y * TS;
    const int col0  = blockIdx.x * TS;
    if (plane >= nplanes) return;
    const float* __restrict__ xp = x   + (size_t)plane * 256 * 256;
    float* __restrict__       op = out + (size_t)plane * 256 * 256;

    // ---- Stage A: stage 68x68 input tile into LDS (b128, conv zero-padding) ----
    // Tile origin col0-4 is 4-aligned, so every float4 is fully in- or out-of-image.
    for (int idx = tid; idx < IS * 17; idx += 256) {          // 68 rows x 17 float4s
        int r = idx / 17, q = idx - r * 17;
        int gr = row0 - 4 + r, gc = col0 - 4 + 4 * q;
        int lo = r * LSTR + 4 * q;
        if (gr >= 0 && gr < 256 && gc >= 0 && gc < 256) {
#ifdef HAS_ASYNC_G2L
            async_g2l_b128(xp + gr * 256 + gc, xs + lo);      // DMA: no VGPR round-trip
#else
            *(v4f*)(xs + lo) = *(const v4f*)(xp + gr * 256 + gc);
#endif
        } else {
            *(v4f*)(xs + lo) = (v4f){0.0f, 0.0f, 0.0f, 0.0f};
        }
    }
    wait_async0();
    __syncthreads();

    // ---- Stage B: horizontal box sums, 4 outputs per b128 pair ----
    for (int idx = tid; idx < IS * 16; idx += 256) {          // 68 rows x 16 float4s
        int r = idx >> 4, q = idx & 15;
        const float* row = xs + r * LSTR + 4 * q;
        v4f e0 = *(const v4f*)(row);
        v4f e1 = *(const v4f*)(row + 4);
        v4f s3, s5;
        s3.x = e0.y + e0.z + e0.w;  s5.x = s3.x + e0.x + e1.x;
        s3.y = e0.z + e0.w + e1.x;  s5.y = s3.y + e0.y + e1.y;
        s3.z = e0.w + e1.x + e1.y;  s5.z = s3.z + e0.z + e1.z;
        s3.w = e1.x + e1.y + e1.z;  s5.w = s3.w + e0.w + e1.w;
        *(v4f*)(h5 + r * LSTR + 4 * q) = s5;
        *(v4f*)(h3 + r * LSTR + 4 * q) = s3;
    }
    __syncthreads();

    // ---- Stage C: vertical banded matmul via V_WMMA_F32_16X16X4_F32 ----
    const int wave = tid >> 5, lane = tid & 31;
    const int half = lane >> 4, ml = lane & 15;   // ml = M (for A/C) or N (for B)

    // Banded A operands, 5 K-chunks of 4.  16x4 f32 A layout:
    // VGPR0 holds K = 2*half, VGPR1 holds K = 2*half+1, M = lane&15.
    v2f a5[5], a3[5];
#pragma unroll
    for (int j = 0; j < 5; ++j) {
        int d0 = 4 * j + 2 * half - ml, d1 = d0 + 1;
        a5[j].x = (d0 >= 0 && d0 <= 4) ? KF : 0.0f;
        a5[j].y = (d1 >= 0 && d1 <= 4) ? KF : 0.0f;
        a3[j].x = (d0 >= 1 && d0 <= 3) ? C3 : 0.0f;
        a3[j].y = (d1 >= 1 && d1 <= 3) ? C3 : 0.0f;
    }

    // 8 waves x 2 tiles cover the 4x4 grid of 16x16 edge tiles
    const int ti0 = wave * 2, ti1 = ti0 + 1;
    const int M0a = (ti0 >> 2) * 16, N0a = (ti0 & 3) * 16;
    const int M0b = (ti1 >> 2) * 16, N0b = (ti1 & 3) * 16;

    // Pre-load C with the delta term 2(H-W)*x (C/D layout: row = M0+v+8*half, col = N0+ml)
    v8f acc0, acc1;
#pragma unroll
    for (int v = 0; v < 8; ++v) {
        acc0[v] = CD * xs[(M0a + v + 8 * half + 2) * LSTR + (N0a + ml + 2)];
        acc1[v] = CD * xs[(M0b + v + 8 * half + 2) * LSTR + (N0b + ml + 2)];
    }
    __syncthreads();   // all xs reads done; edge buffer overlays xs below

#pragma unroll
    for (int t = 0; t < 2; ++t) {
        const int M0 = t ? M0b : M0a, N0 = t ? N0b : N0a;
        v8f c = t ? acc1 : acc0;
#pragma unroll
        for (int j = 0; j < 5; ++j) {        // edge += F * box5(vertical)
            int kr = M0 + 4 * j + 2 * half;  // B layout: VGPR0 row 2*half, VGPR1 row 2*half+1
            v2f b;
            b.x = h5[kr * LSTR + N0 + ml];
            b.y = h5[(kr + 1) * LSTR + N0 + ml];
            c = __builtin_amdgcn_wmma_f32_16x16x4_f32(false, a5[j], false, b,
                                                      (short)0, c, false, false);
        }
#pragma unroll
        for (int j = 0; j < 5; ++j) {        // edge += (2W-F) * box3(vertical)
            int kr = M0 + 4 * j + 2 * half;
            v2f b;
            b.x = h3[kr * LSTR + N0 + ml];
            b.y = h3[(kr + 1) * LSTR + N0 + ml];
            c = __builtin_amdgcn_wmma_f32_16x16x4_f32(false, a3[j], false, b,
                                                      (short)0, c, false, false);
        }
        // Store edge; positions outside the image get -inf (maxpool ignores them)
#pragma unroll
        for (int v = 0; v < 8; ++v) {
            int er = M0 + v + 8 * half, ec = N0 + ml;
            int gr = row0 - 2 + er, gc = col0 - 2 + ec;
            float val = (gr >= 0 && gr < 256 && gc >= 0 && gc < 256) ? c[v]
                                                                     : -__builtin_inff();
            xs[er * LSTR + ec] = val;   // edge overlay on xs
        }
    }
    __syncthreads();

    // ---- Stage D: separable 5x5 maxpool + abs, all b128 ----
    float* edge = xs;
    float* rmax = h5;   // overlay
    for (int idx = tid; idx < ES * 15; idx += 256) {   // pass 1: row max5, 64 x 15 float4s
        int r = idx / 15, q = idx - r * 15;
        const float* e = edge + r * LSTR + 4 * q;
        v4f e0 = *(const v4f*)(e);
        v4f e1 = *(const v4f*)(e + 4);
        v4f m;
        m.x = max5(e0.x, e0.y, e0.z, e0.w, e1.x);
        m.y = max5(e0.y, e0.z, e0.w, e1.x, e1.y);
        m.z = max5(e0.z, e0.w, e1.x, e1.y, e1.z);
        m.w = max5(e0.w, e1.x, e1.y, e1.z, e1.w);
        *(v4f*)(rmax + r * LSTR + 4 * q) = m;
    }
    __syncthreads();
    for (int idx = tid; idx < TS * 15; idx += 256) {   // pass 2: col max5, abs, b128 store
        int i = idx / 15, q = idx - i * 15;
        const float* rm = rmax + i * LSTR + 4 * q;
        v4f r0 = *(const v4f*)(rm);
        v4f r1 = *(const v4f*)(rm + LSTR);
        v4f r2 = *(const v4f*)(rm + 2 * LSTR);
        v4f r3 = *(const v4f*)(rm + 3 * LSTR);
        v4f r4 = *(const v4f*)(rm + 4 * LSTR);
        v4f m;
        m.x = fabsf(max5(r0.x, r1.x, r2.x, r3.x, r4.x));
        m.y = fabsf(max5(r0.y, r1.y, r2.y, r3.y, r4.y));
        m.z = fabsf(max5(r0.z, r1.z, r2.z, r3.z, r4.z));
        m.w = fabsf(max5(r0.w, r1.w, r2.w, r3.w, r4.w));
        int gr = row0 + i, gc = col0 + 4 * q;          // 4-aligned: fully in or out
        if (gr < 256 && gc < 256)
            *(v4f*)(op + gr * 256 + gc) = m;
    }
}

extern "C" void kernel_launch(void* const* d_in, const int* in_sizes, int n_in,
                              void* d_out, int out_size, void* d_ws, size_t ws_size,
                              hipStream_t stream) {
    const float* x = (const float*)d_in[0];
    float* out = (float*)d_out;
    int nplanes = in_sizes[0] / (256 * 256);   // B*C = 512 for [8,64,256,256]
    dim3 grid((256 + TS - 1) / TS, (256 + TS - 1) / TS, nplanes);  // 5 x 5 x 512
    dim3 block(256);
    hipLaunchKernelGGL(edge_pool_kernel, grid, block, 0, stream, x, out, nplanes);
}